// GHM_75127567941751
// MI455X (gfx1250) — compile-verified
//
#include <hip/hip_runtime.h>
#include <stdint.h>

// ---------------------------------------------------------------------------
// GHM loss, single data pass:
//   sum_i w[idx_i]*bce_i == sum_b bin_w[b] * (sum over elements in bin b of bce)
// Pass 1 (ghm_main): per-lane accumulate totBce, and for k=1..9:
//   geC[k] = #elements with g >= k/10   (exact u32)
//   geB[k] = sum of bce over elements with g >= k/10
// Pass 2 (ghm_finalize): diff the ">=" sums into per-bin counts/bce sums,
//   apply momentum bin weights, emit scalar.
// No atomics anywhere -> bitwise deterministic across graph replays.
// ---------------------------------------------------------------------------

#define GHM_THREADS 256

// -------- optional CDNA5 async global->LDS streaming path ------------------
#ifndef GHM_HAVE_ASYNC
#if defined(__has_builtin)
#if __has_builtin(__builtin_amdgcn_global_load_async_to_lds_b128) && \
    __has_builtin(__builtin_amdgcn_s_wait_asynccnt)
#define GHM_HAVE_ASYNC 1
#endif
#endif
#endif
#ifndef GHM_HAVE_ASYNC
#define GHM_HAVE_ASYNC 0
#endif

#if GHM_HAVE_ASYNC
typedef int ghm_v4i __attribute__((ext_vector_type(4)));
typedef __attribute__((address_space(1))) ghm_v4i ghm_gbl_v4i;  // global (AS1)
typedef __attribute__((address_space(3))) ghm_v4i ghm_lds_v4i;  // LDS (AS3)
__device__ __forceinline__ void ghm_async_cp16(const void* gsrc, void* ldst) {
  // global_load_async_to_lds_b128: 16B per lane, tracked by ASYNCcnt
  __builtin_amdgcn_global_load_async_to_lds_b128(
      (ghm_gbl_v4i*)gsrc, (ghm_lds_v4i*)ldst, /*offset=*/0, /*cpol=*/0);
}
#endif

// -------- per-element math -------------------------------------------------
__device__ __forceinline__ void ghm_elem(float x, int ti, float& totB,
                                         float (&geB)[9], unsigned (&geC)[9]) {
  const float ax  = __builtin_fabsf(x);
  const float e   = __expf(-ax);                       // exp(-|x|)
  const float u   = __builtin_amdgcn_rcpf(1.0f + e);   // 1/(1+exp(-|x|))
  const float sig = (x >= 0.0f) ? u : 1.0f - u;        // sigmoid(x)
  const float tf  = (float)ti;
  const float g   = __builtin_fabsf(sig - tf);
  // stable BCE-with-logits: max(x,0) - x*t + log1p(exp(-|x|))
  const float bce = __builtin_fmaxf(x, 0.0f) - x * tf + __logf(1.0f + e);
  totB += bce;
  const float B[9] = {0.1f, 0.2f, 0.3f, 0.4f, 0.5f, 0.6f, 0.7f, 0.8f, 0.9f};
#pragma unroll
  for (int k = 0; k < 9; ++k) {
    const bool c = (g >= B[k]);   // matches searchsorted(border, g, 'right')-1
    geC[k] += (unsigned)c;
    geB[k] += c ? bce : 0.0f;
  }
}

// -------- pass 1: streaming accumulate ------------------------------------
__global__ __launch_bounds__(GHM_THREADS) void ghm_main(
    const float4* __restrict__ pred4, const int4* __restrict__ tgt4,
    float* __restrict__ bcePart, unsigned* __restrict__ cntPart,
    unsigned n, unsigned n4) {
  const unsigned tid    = threadIdx.x;
  const unsigned nb     = gridDim.x;
  const unsigned stride = nb * GHM_THREADS;  // in float4 units
  unsigned i = blockIdx.x * GHM_THREADS + tid;

  float    totB = 0.0f;
  float    geB[9];
  unsigned geC[9];
#pragma unroll
  for (int k = 0; k < 9; ++k) { geB[k] = 0.0f; geC[k] = 0u; }

#if GHM_HAVE_ASYNC
  // Per-wave double-buffered async global->LDS streaming (ASYNCcnt paced).
  // Each lane copies its own 16B of pred and target; a wave only ever reads
  // the LDS it wrote itself, so no __syncthreads needed in the loop.
  __shared__ float4 sP[2][GHM_THREADS];
  __shared__ int4   sT[2][GHM_THREADS];
  const unsigned nFull = n4 / stride;  // uniform across the block
  if (nFull > 0) {
    ghm_async_cp16(pred4 + i, &sP[0][tid]);
    ghm_async_cp16(tgt4 + i, &sT[0][tid]);
  }
  for (unsigned it = 0; it < nFull; ++it) {
    const unsigned nxt = it + 1u;
    if (nxt < nFull) {
      const unsigned gi = i + nxt * stride;
      ghm_async_cp16(pred4 + gi, &sP[nxt & 1u][tid]);
      ghm_async_cp16(tgt4 + gi, &sT[nxt & 1u][tid]);
      __builtin_amdgcn_s_wait_asynccnt(2);  // oldest pair (iter `it`) done
    } else {
      __builtin_amdgcn_s_wait_asynccnt(0);
    }
    const float4 p = sP[it & 1u][tid];  // ds_load_b128
    const int4   t = sT[it & 1u][tid];
    ghm_elem(p.x, t.x, totB, geB, geC);
    ghm_elem(p.y, t.y, totB, geB, geC);
    ghm_elem(p.z, t.z, totB, geB, geC);
    ghm_elem(p.w, t.w, totB, geB, geC);
  }
  i += nFull * stride;
#endif
  for (; i < n4; i += stride) {
#if !GHM_HAVE_ASYNC
    __builtin_prefetch((const void*)(pred4 + i + stride), 0, 1);  // global_prefetch_b8
    __builtin_prefetch((const void*)(tgt4 + i + stride), 0, 1);
#endif
    const float4 p = pred4[i];  // global_load_b128
    const int4   t = tgt4[i];
    ghm_elem(p.x, t.x, totB, geB, geC);
    ghm_elem(p.y, t.y, totB, geB, geC);
    ghm_elem(p.z, t.z, totB, geB, geC);
    ghm_elem(p.w, t.w, totB, geB, geC);
  }
  // scalar tail (n not multiple of 4) handled by one thread
  if (blockIdx.x == 0 && tid == 0) {
    const float* ps = (const float*)pred4;
    const int*   ts = (const int*)tgt4;
    for (unsigned r = n4 * 4u; r < n; ++r) ghm_elem(ps[r], ts[r], totB, geB, geC);
  }

  // wave32 shuffle reduction of all 19 accumulators
#pragma unroll
  for (int off = 16; off > 0; off >>= 1) {
    totB += __shfl_down(totB, (unsigned)off, 32);
#pragma unroll
    for (int k = 0; k < 9; ++k) {
      geB[k] += __shfl_down(geB[k], (unsigned)off, 32);
      geC[k] += __shfl_down(geC[k], (unsigned)off, 32);
    }
  }
  __shared__ float    sF[10][8];
  __shared__ unsigned sU[9][8];
  const unsigned wid = tid >> 5, lane = tid & 31u;
  if (lane == 0) {
    sF[0][wid] = totB;
#pragma unroll
    for (int k = 0; k < 9; ++k) { sF[k + 1][wid] = geB[k]; sU[k][wid] = geC[k]; }
  }
  __syncthreads();
  if (tid < 10u) {
    float s = 0.0f;
#pragma unroll
    for (int w = 0; w < 8; ++w) s += sF[tid][w];   // fixed order: deterministic
    bcePart[tid * nb + blockIdx.x] = s;
  } else if (tid >= 16u && tid < 25u) {
    const unsigned j = tid - 16u;
    unsigned s = 0u;
#pragma unroll
    for (int w = 0; w < 8; ++w) s += sU[j][w];
    cntPart[j * nb + blockIdx.x] = s;
  }
}

// -------- pass 2: finalize (1 block) ---------------------------------------
__global__ __launch_bounds__(GHM_THREADS) void ghm_finalize(
    const float* __restrict__ bcePart, const unsigned* __restrict__ cntPart,
    const float* __restrict__ acc_sum, float* __restrict__ out,
    unsigned n, unsigned nb) {
  const unsigned tid = threadIdx.x;
  float    locF[10];
  unsigned locU[9];
#pragma unroll
  for (int j = 0; j < 10; ++j) locF[j] = 0.0f;
#pragma unroll
  for (int j = 0; j < 9; ++j) locU[j] = 0u;
  for (unsigned i = tid; i < nb; i += GHM_THREADS) {
#pragma unroll
    for (int j = 0; j < 10; ++j) locF[j] += bcePart[(unsigned)j * nb + i];
#pragma unroll
    for (int j = 0; j < 9; ++j) locU[j] += cntPart[(unsigned)j * nb + i];
  }
#pragma unroll
  for (int off = 16; off > 0; off >>= 1) {
#pragma unroll
    for (int j = 0; j < 10; ++j) locF[j] += __shfl_down(locF[j], (unsigned)off, 32);
#pragma unroll
    for (int j = 0; j < 9; ++j) locU[j] += __shfl_down(locU[j], (unsigned)off, 32);
  }
  __shared__ float    sF[10][8];
  __shared__ unsigned sU[9][8];
  const unsigned wid = tid >> 5, lane = tid & 31u;
  if (lane == 0) {
#pragma unroll
    for (int j = 0; j < 10; ++j) sF[j][wid] = locF[j];
#pragma unroll
    for (int j = 0; j < 9; ++j) sU[j][wid] = locU[j];
  }
  __syncthreads();
  if (tid == 0) {
    float    F[10];
    unsigned U[9];
#pragma unroll
    for (int j = 0; j < 10; ++j) {
      float s = 0.0f;
#pragma unroll
      for (int w = 0; w < 8; ++w) s += sF[j][w];
      F[j] = s;
    }
#pragma unroll
    for (int j = 0; j < 9; ++j) {
      unsigned s = 0u;
#pragma unroll
      for (int w = 0; w < 8; ++w) s += sU[j][w];
      U[j] = s;
    }
    // per-bin counts (exact) and per-bin bce sums from ">=" sums
    unsigned cnts[10];
    float    binB[10];
    cnts[0] = n - U[0];
    binB[0] = F[0] - F[1];
#pragma unroll
    for (int b = 1; b < 9; ++b) {
      cnts[b] = U[b - 1] - U[b];
      binB[b] = F[b] - F[b + 1];
    }
    cnts[9] = U[8];
    binB[9] = F[9];
    // momentum bin weights (matches reference)
    float bw[10];
    int   nn = 0;
    const float tot = (float)n;
#pragma unroll
    for (int b = 0; b < 10; ++b) {
      if (cnts[b] > 0u) {
        const float accn = 0.9f * acc_sum[b] + 0.1f * (float)cnts[b];
        bw[b] = tot / accn;
        ++nn;
      } else {
        bw[b] = 0.0f;
      }
    }
    if (nn > 0) {
      const float nf = (float)nn;
#pragma unroll
      for (int b = 0; b < 10; ++b) bw[b] = bw[b] / nf;
    }
    float r = 0.0f;
#pragma unroll
    for (int b = 0; b < 10; ++b) r += bw[b] * binB[b];
    out[0] = r / tot;
  }
}

// -------- host-side launcher -----------------------------------------------
extern "C" void kernel_launch(void* const* d_in, const int* in_sizes, int n_in,
                              void* d_out, int out_size, void* d_ws, size_t ws_size,
                              hipStream_t stream) {
  const float4* pred4   = (const float4*)d_in[0];
  const int4*   tgt4    = (const int4*)d_in[1];
  const float*  acc_sum = (const float*)d_in[2];
  float*        out     = (float*)d_out;
  const unsigned n  = (unsigned)in_sizes[0];
  const unsigned n4 = n / 4u;

  // pick block count so per-block partials fit in workspace
  unsigned nb = 1024u;
  while (nb > 1u && ws_size < (size_t)19u * nb * sizeof(float)) nb >>= 1;

  float*    bcePart = (float*)d_ws;                                  // [10][nb]
  unsigned* cntPart = (unsigned*)((char*)d_ws + (size_t)10u * nb * 4u);  // [9][nb]

  ghm_main<<<nb, GHM_THREADS, 0, stream>>>(pred4, tgt4, bcePart, cntPart, n, n4);
  ghm_finalize<<<1, GHM_THREADS, 0, stream>>>(bcePart, cntPart, acc_sum, out, n, nb);
}